// EdgeEncoder_8495445311732
// MI455X (gfx1250) — compile-verified
//
#include <hip/hip_runtime.h>
#include <hip/hip_bf16.h>

// Problem constants (from reference): N=512, L=5, H=8, D=16, E=10000
#define PN   512
#define PL   5
#define PH   8
#define PD   16
#define PNN  (PN * PN)          // 262144 pairs
#define TILES (PNN / 16)        // 16384 tiles of 16 pairs
#define WAVES_PER_BLOCK 8
#define BLOCKS (TILES / WAVES_PER_BLOCK)  // 2048

typedef float v2f __attribute__((ext_vector_type(2)));
typedef float v8f __attribute__((ext_vector_type(8)));

__global__ __launch_bounds__(256) void edge_encoder_wmma(
    const float* __restrict__ feats,   // [E, 16] f32
    const float* __restrict__ W,       // [6, 128] f32 (rows 1..5 used)
    const int*   __restrict__ sp,      // [N*N, 5] int32, -1 = padding
    float* __restrict__ out)           // [8, N*N] f32
{
    // B matrix staged in LDS, h-padded to 16 columns of zeros:
    // Blds[k][n] with k = l*16 + d (K=80), n = h (n>=8 -> 0)
    __shared__ float Blds[80 * 16];
    __shared__ float stage[WAVES_PER_BLOCK][16 * 8]; // [pair][h] per wave
    __shared__ float nrl[WAVES_PER_BLOCK][16];       // 1/n_ij per pair

    const int tid = threadIdx.x;

    for (int t = tid; t < 80 * 16; t += 256) {
        const int k = t >> 4;
        const int n = t & 15;
        float v = 0.0f;
        if (n < 8)
            v = W[(1 + (k >> 4)) * (PH * PD) + n * PD + (k & 15)];
        Blds[t] = v;
    }
    __syncthreads();

    const int lane = tid & 31;
    const int wv   = tid >> 5;
    const int hi   = lane >> 4;        // 0: lanes 0-15, 1: lanes 16-31
    const int ln16 = lane & 15;

    const int tile = blockIdx.x * WAVES_PER_BLOCK + wv;
    const int p0   = tile * 16;        // first flat pair index of this tile
    const int pj   = p0 + ln16;        // pair handled by this lane (dup in both halves)

    // Load 5 path-edge indices for this pair; build clipped indices + masks.
    int   idx[PL];
    float msk[PL];
    int   cnt = 0;
    #pragma unroll
    for (int l = 0; l < PL; ++l) {
        const int v = sp[pj * PL + l];
        const bool valid = (v != -1);
        cnt   += valid ? 1 : 0;
        idx[l] = valid ? v : 0;
        msk[l] = valid ? 1.0f : 0.0f;
    }
    const float nrecip = 1.0f / (float)(cnt > 0 ? cnt : 1);
    if (hi == 0) nrl[wv][ln16] = nrecip;

    // Accumulate D[pair][h] = sum_{l,d} maskedFeat[pair][l][d] * W[l][h][d]
    // over 20 x V_WMMA_F32_16X16X4_F32 steps (K = 80 total).
    v8f c = {};
    #pragma unroll
    for (int l = 0; l < PL; ++l) {
        const float* fbase = feats + (size_t)idx[l] * PD;
        const float  m     = msk[l];
        #pragma unroll
        for (int t4 = 0; t4 < 4; ++t4) {
            const int s     = l * 4 + t4;       // WMMA step, K-chunk base = 4s
            const int dbase = t4 * 4 + hi * 2;  // A: K local {0,1} / {2,3}
            v2f a;
            a.x = m * fbase[dbase];
            a.y = m * fbase[dbase + 1];
            const int kb = s * 4 + hi * 2;      // B: K rows per lane-half
            v2f b;
            b.x = Blds[kb * 16 + ln16];
            b.y = Blds[(kb + 1) * 16 + ln16];
            // (neg_a, A, neg_b, B, c_mod, C, reuse_a, reuse_b)
            c = __builtin_amdgcn_wmma_f32_16x16x4_f32(
                    false, a, false, b, (short)0, c, false, false);
        }
    }

    // C/D layout: VGPR r, lane -> (pair = r + 8*hi, h = lane&15). Valid h < 8.
    if (ln16 < 8) {
        #pragma unroll
        for (int r = 0; r < 8; ++r) {
            const int pm = r + 8 * hi;
            stage[wv][pm * 8 + ln16] = c[r];
        }
    }
    // LDS is in-order within a wave; fence the transpose explicitly (CDNA5
    // split dependency counter) and stop the compiler reordering across it.
    asm volatile("s_wait_dscnt 0x0" ::: "memory");

    // Coalesced stores: per iteration two h-rows of 16 contiguous floats.
    #pragma unroll
    for (int q = 0; q < 4; ++q) {
        const int hh  = 2 * q + hi;
        const float v = stage[wv][ln16 * 8 + hh] * nrl[wv][ln16];
        out[(size_t)hh * PNN + p0 + ln16] = v;
    }
}

extern "C" void kernel_launch(void* const* d_in, const int* in_sizes, int n_in,
                              void* d_out, int out_size, void* d_ws, size_t ws_size,
                              hipStream_t stream) {
    (void)in_sizes; (void)n_in; (void)out_size; (void)d_ws; (void)ws_size;
    const float* feats = (const float*)d_in[0];   // edge_features_s [E*D]
    const float* W     = (const float*)d_in[1];   // edge_weights [(L+1)*H*D]
    const int*   sp    = (const int*)d_in[2];     // shortest_path_edges [N*N*L]
    float*       out   = (float*)d_out;           // [H, N, N]

    edge_encoder_wmma<<<BLOCKS, 256, 0, stream>>>(feats, W, sp, out);
}